// PatchExtractor2d_28750511079629
// MI455X (gfx1250) — compile-verified
//
#include <hip/hip_runtime.h>

// ---- problem constants (baked from the reference) ----
#define B_   8
#define C_   64
#define H_   224
#define W_   224
#define HO_  222
#define WO_  222
#define C2_  576          // 9 * C_
#define TH_  16           // output rows per block
#define RB_  14           // ceil(HO_ / TH_)
#define THREADS 256

typedef float v2f __attribute__((ext_vector_type(2)));
typedef float v4f __attribute__((ext_vector_type(4)));
typedef int   v4i __attribute__((ext_vector_type(4)));

#if defined(__AMDGCN__) && defined(__gfx1250__) && \
    __has_builtin(__builtin_amdgcn_global_load_async_to_lds_b128)
#define USE_ASYNC_LDS 1
#else
#define USE_ASYNC_LDS 0
#endif

// Builtin signature (from the round-1 diagnostic): param 0 is
// 'int4 __device__*' i.e. addrspace(1) int4*; use typed AS casts.
#define AS_GLOBAL_V4I(p) ((__attribute__((address_space(1))) v4i*)(p))
#define AS_LDS_V4I(p)    ((__attribute__((address_space(3))) v4i*)(p))

__global__ __launch_bounds__(THREADS)
void patch_extract_kernel(const float* __restrict__ x, float* __restrict__ out) {
    __shared__ float smem[(TH_ + 2) * W_];   // 18*224*4 = 16128 B

    const int tid = threadIdx.x;
    const int r   = blockIdx.x * TH_;        // first output row of this tile
    const int c   = blockIdx.y;              // input channel
    const int b   = blockIdx.z;              // batch

    const int rows_out = (HO_ - r) < TH_ ? (HO_ - r) : TH_;
    const int rows_in  = rows_out + 2;       // halo of K-1 = 2 rows

    // Input span: rows [r, r+rows_in) of plane (b,c) — contiguous, 16B aligned
    const float* src = x + (((size_t)(b * C_ + c)) * H_ + (size_t)r) * W_;
    const int chunks = rows_in * (W_ / 4);   // number of 16-byte chunks

#if USE_ASYNC_LDS
    // CDNA5 async DMA: global -> LDS without touching VGPRs (ASYNCcnt-tracked)
    for (int ci = tid; ci < chunks; ci += THREADS) {
        __builtin_amdgcn_global_load_async_to_lds_b128(
            AS_GLOBAL_V4I(src + ci * 4), AS_LDS_V4I(&smem[ci * 4]), 0, 0);
    }
#if __has_builtin(__builtin_amdgcn_s_wait_asynccnt)
    __builtin_amdgcn_s_wait_asynccnt(0);
#else
    asm volatile("s_wait_asynccnt 0" ::: "memory");
#endif
#else
    // Fallback: vectorized load through VGPRs into LDS
    for (int ci = tid; ci < chunks; ci += THREADS) {
        v4f v = *(const v4f*)(src + ci * 4);
        *(v4f*)(&smem[ci * 4]) = v;
    }
#endif
    __syncthreads();

    // Fan out: 9 shifted 222-wide crops -> 9 output channel planes.
    // Output row = 222 floats = 888 B = 111 x float2, always 8B aligned.
    const size_t planeRowBase = ((size_t)b * C2_ + (size_t)9 * c) * HO_;
    for (int jl = 0; jl < 9; ++jl) {
        const int j = jl / 3;
        const int l = jl - 3 * j;
        float* dst = out + (planeRowBase + (size_t)jl * HO_ + (size_t)r) * WO_;
        const float* sbase = smem + j * W_ + l;
        const int elems = rows_out * (WO_ / 2);   // float2 elements in this tile
        for (int idx = tid; idx < elems; idx += THREADS) {
            const int row = idx / 111;
            const int w2  = (idx - row * 111) * 2;
            const float* s = sbase + row * W_ + w2;
            v2f v;
            v.x = s[0];
            v.y = s[1];
            // NT store: keep the 908MB write stream from evicting the
            // L2-resident (103MB < 192MB) input tensor.
            __builtin_nontemporal_store(v, (v2f*)(dst + (size_t)row * WO_ + w2));
        }
    }
}

extern "C" void kernel_launch(void* const* d_in, const int* in_sizes, int n_in,
                              void* d_out, int out_size, void* d_ws, size_t ws_size,
                              hipStream_t stream) {
    const float* x = (const float*)d_in[0];
    // d_in[1] is the one-hot im2col weight; its mapping (cout = 9*cin+3*j+l)
    // is baked into the kernel's addressing, so it is not read.
    (void)in_sizes; (void)n_in; (void)d_ws; (void)ws_size; (void)out_size;
    float* out = (float*)d_out;

    dim3 grid(RB_, C_, B_);   // 14 x 64 x 8 = 7168 blocks, 256 threads (8 waves) each
    patch_extract_kernel<<<grid, THREADS, 0, stream>>>(x, out);
}